// RelTransformerV_29721173688652
// MI455X (gfx1250) — compile-verified
//
#include <hip/hip_runtime.h>

typedef __attribute__((ext_vector_type(2))) float v2f;
typedef __attribute__((ext_vector_type(8))) float v8f;
typedef __attribute__((ext_vector_type(4))) unsigned int v4u;
typedef __attribute__((ext_vector_type(4))) int v4i;
typedef __attribute__((ext_vector_type(8))) int v8i;

#define WMMA_F32(A_, B_, C_) \
  __builtin_amdgcn_wmma_f32_16x16x4_f32(false, (A_), false, (B_), (short)0, (C_), false, false)

constexpr int Bsz = 32;   // batch
constexpr int Tn  = 128;  // sequence
constexpr int Kd  = 32;   // feature dim per head
constexpr int Hn  = 4;    // heads
constexpr int HK  = 128;  // H*K
constexpr int HID = 256;  // MLP hidden
constexpr float SCALE = 0.17677669529663687f; // 1/sqrt(32)

// ---------------------------------------------------------------------------
// TDM: issue a 2-D tensor_load_to_lds.  Descriptor per CDNA5 ISA ch.8:
//   group0: count=1 | lds_addr | global_addr(57b) | type=2
//   group1: data_size=4B, tensor_dim0/1, tile_dim0/1, tensor_dim0_stride
// dim0/tile0 = contiguous extent (elements), dim1/tile1 = rows,
// stride0 = row stride in elements (lets us slice one head out of HK=128).
// This toolchain's builtin takes 6 operands (extra int32x8 group, zeroed).
// ---------------------------------------------------------------------------
__device__ __forceinline__ void tdm_load_2d(unsigned lds_off, const void* gptr,
                                            unsigned dim0, unsigned dim1,
                                            unsigned stride0,
                                            unsigned tile0, unsigned tile1) {
  unsigned long long ga = (unsigned long long)(size_t)gptr;
  v4u g0;
  g0[0] = 1u;                                              // count=1 (valid)
  g0[1] = lds_off;                                         // LDS byte address
  g0[2] = (unsigned)(ga & 0xffffffffull);                  // global addr lo
  g0[3] = (unsigned)((ga >> 32) & 0x1ffffffull) | (2u << 30); // addr hi | type=2
  v8i g1;
  g1[0] = (int)(2u << 16);                                 // data_size = 4 bytes
  g1[1] = (int)((dim0 & 0xffffu) << 16);                   // tensor_dim0 [79:48]
  g1[2] = (int)((dim0 >> 16) | ((dim1 & 0xffffu) << 16));  // dim0 hi | tensor_dim1 lo
  g1[3] = (int)((dim1 >> 16) | (tile0 << 16));             // dim1 hi | tile_dim0
  g1[4] = (int)(tile1 & 0xffffu);                          // tile_dim1 (tile_dim2=0)
  g1[5] = (int)stride0;                                    // tensor_dim0_stride lo
  g1[6] = 0;
  g1[7] = 0;
  v4i gz4 = {0, 0, 0, 0};
  v8i gz8 = {0, 0, 0, 0, 0, 0, 0, 0};
  __builtin_amdgcn_tensor_load_to_lds(g0, g1, gz4, gz4, gz8, 0);
}

// ---------------------------------------------------------------------------
// Kernel 1: Q/Kf/Vf = state @ {Wq,Wk,Wv}.  grid (B, 3), 256 threads (8 waves).
// TDM stages state tile (128x32) and W (32x128) into LDS, WMMA feeds from LDS.
// ---------------------------------------------------------------------------
__global__ void proj_kernel(const float* __restrict__ state,
                            const float* __restrict__ Wq,
                            const float* __restrict__ Wk,
                            const float* __restrict__ Wv,
                            float* __restrict__ Q,
                            float* __restrict__ Kf,
                            float* __restrict__ Vf) {
  __shared__ float sls[Tn][Kd];   // 16 KB state tile
  __shared__ float wls[Kd][HK];   // 16 KB weight
  const int b = blockIdx.x;
  const int w = blockIdx.y;
  const float* W = (w == 0) ? Wq : (w == 1) ? Wk : Wv;   // (32, 128) row-major
  float* Out = (w == 0) ? Q : (w == 1) ? Kf : Vf;        // (B, T, 128)

  const int tid   = threadIdx.x;
  const int wave  = tid >> 5;
  const int lane  = tid & 31;
  const int l16   = lane & 15;
  const int lhalf = lane >> 4;
  const int lh2   = lhalf << 1;

  const float* Sb = state + (size_t)b * Tn * Kd;
  float*       Ob = Out   + (size_t)b * Tn * HK;

  if (wave == 0) {
    tdm_load_2d((unsigned)(size_t)&sls[0][0], Sb, Kd, Tn, Kd, Kd, Tn);
    tdm_load_2d((unsigned)(size_t)&wls[0][0], W,  HK, Kd, HK, HK, Kd);
    __builtin_amdgcn_s_wait_tensorcnt(0);
  }
  __syncthreads();

  const int mt = wave;
  const int arow = mt * 16 + l16;
  v2f afrag[8];
#pragma unroll
  for (int kk = 0; kk < 8; ++kk)
    afrag[kk] = *(const v2f*)&sls[arow][kk * 4 + lh2];

  for (int nt = 0; nt < 8; ++nt) {
    const int col = nt * 16 + l16;
    v2f bf[8];
#pragma unroll
    for (int kk = 0; kk < 8; ++kk) {
      const int k0 = kk * 4 + lh2;
      bf[kk][0] = wls[k0 + 0][col];
      bf[kk][1] = wls[k0 + 1][col];
    }
    v8f c = {};
#pragma unroll
    for (int kk = 0; kk < 8; ++kk) c = WMMA_F32(afrag[kk], bf[kk], c);
#pragma unroll
    for (int r = 0; r < 8; ++r) {
      const int row = mt * 16 + r + (lhalf << 3);
      Ob[row * HK + col] = c[r];
    }
  }
}

// ---------------------------------------------------------------------------
// Kernel 2: per (b,h) attention.  grid (B, H), 256 threads.
// TDM stages strided Q/K/V head slices (tensor_dim0=32, stride=128) into LDS.
//   A = Q_h @ Kf_h^T; diag-masked softmax; X = w @ Vf_h - Vf_h.
// ---------------------------------------------------------------------------
__global__ void attn_kernel(const float* __restrict__ Q,
                            const float* __restrict__ Kf,
                            const float* __restrict__ Vf,
                            float* __restrict__ X) {
  __shared__ float smem[Tn][Tn];  // 64 KB attention matrix
  __shared__ float qls[Tn][Kd];   // 16 KB
  __shared__ float kls[Tn][Kd];   // 16 KB
  __shared__ float vls[Tn][Kd];   // 16 KB
  const int b = blockIdx.x, h = blockIdx.y;
  const float* Qb = Q  + (size_t)b * Tn * HK + h * Kd;
  const float* Kb = Kf + (size_t)b * Tn * HK + h * Kd;
  const float* Vb = Vf + (size_t)b * Tn * HK + h * Kd;

  const int tid   = threadIdx.x;
  const int wave  = tid >> 5;
  const int lane  = tid & 31;
  const int l16   = lane & 15;
  const int lhalf = lane >> 4;
  const int lh2   = lhalf << 1;
  const int mt    = wave;

  if (wave == 0) {
    tdm_load_2d((unsigned)(size_t)&qls[0][0], Qb, Kd, Tn, HK, Kd, Tn);
    tdm_load_2d((unsigned)(size_t)&kls[0][0], Kb, Kd, Tn, HK, Kd, Tn);
    tdm_load_2d((unsigned)(size_t)&vls[0][0], Vb, Kd, Tn, HK, Kd, Tn);
    __builtin_amdgcn_s_wait_tensorcnt(0);
  }
  __syncthreads();

  // ---- scores: A = Q_h @ Kf_h^T ----
  v2f afrag[8];
#pragma unroll
  for (int kk = 0; kk < 8; ++kk)
    afrag[kk] = *(const v2f*)&qls[mt * 16 + l16][kk * 4 + lh2];

  for (int nt = 0; nt < 8; ++nt) {
    v2f bf[8];
#pragma unroll
    for (int kk = 0; kk < 8; ++kk)
      bf[kk] = *(const v2f*)&kls[nt * 16 + l16][kk * 4 + lh2];
    v8f c = {};
#pragma unroll
    for (int kk = 0; kk < 8; ++kk) c = WMMA_F32(afrag[kk], bf[kk], c);
#pragma unroll
    for (int r = 0; r < 8; ++r) {
      const int i = mt * 16 + r + (lhalf << 3);
      const int j = nt * 16 + l16;
      float v = c[r] * SCALE;
      if (i == j) v = -1e30f;                  // mask self (j != i softmax)
      smem[i][j] = v;
    }
  }
  __syncthreads();

  // ---- row softmax (threads 0..127, one row each) ----
  if (tid < Tn) {
    float m = -1e30f;
    for (int j = 0; j < Tn; ++j) m = fmaxf(m, smem[tid][j]);
    float s = 0.f;
    for (int j = 0; j < Tn; ++j) {
      float e = __expf(smem[tid][j] - m);
      smem[tid][j] = e;
      s += e;
    }
    const float inv = 1.0f / s;
    for (int j = 0; j < Tn; ++j) smem[tid][j] *= inv;
  }
  __syncthreads();

  // ---- X_h = w @ Vf_h - Vf_h ----
  for (int nt = 0; nt < 2; ++nt) {
    const int n = nt * 16 + l16;
    v8f c = {};
#pragma unroll
    for (int kk = 0; kk < 32; ++kk) {
      const int k0 = kk * 4 + lh2;
      v2f a = *(const v2f*)&smem[mt * 16 + l16][k0];
      v2f bb;
      bb[0] = vls[k0 + 0][n];
      bb[1] = vls[k0 + 1][n];
      c = WMMA_F32(a, bb, c);
    }
#pragma unroll
    for (int r = 0; r < 8; ++r) {
      const int i = mt * 16 + r + (lhalf << 3);
      X[(size_t)b * Tn * HK + i * HK + h * Kd + n] = c[r] - vls[i][n];
    }
  }
}

// ---------------------------------------------------------------------------
// Kernel 3: fused MLP.  grid 64 blocks x 64 rows; activations in LDS.
// B fragments double-buffered in registers (8-step batches) so weight loads
// overlap the WMMA chain instead of a full s_wait per step.
//   in = [X | state] (160) -> relu 256 -> relu 256 -> 1
// ---------------------------------------------------------------------------
__global__ void mlp_kernel(const float* __restrict__ X,
                           const float* __restrict__ state,
                           const float* __restrict__ W1, const float* __restrict__ b1,
                           const float* __restrict__ W2, const float* __restrict__ b2,
                           const float* __restrict__ Wo, const float* __restrict__ bo,
                           float* __restrict__ out) {
  __shared__ float act1[64][HID];   // 64 KB
  __shared__ float act2[64][HID];   // 64 KB
  const int row0  = blockIdx.x * 64;
  const int tid   = threadIdx.x;
  const int wave  = tid >> 5;
  const int lane  = tid & 31;
  const int l16   = lane & 15;
  const int lhalf = lane >> 4;
  const int lh2   = lhalf << 1;
  const int mt    = wave & 3;             // 4 M tiles of 16 rows
  const int nt0   = (wave >> 2) * 8;      // each wave covers 8 N tiles

  const int arow = row0 + mt * 16 + l16;

  // ---- layer 1: (64 x 160) @ W1 (160 x 256); K = 5 batches of 8 steps ----
  for (int nt = nt0; nt < nt0 + 8; ++nt) {
    const int col = nt * 16 + l16;
    v2f bf[2][8];
#pragma unroll
    for (int i = 0; i < 8; ++i) {
      const int k0 = i * 4 + lh2;
      bf[0][i][0] = W1[(k0 + 0) * HID + col];
      bf[0][i][1] = W1[(k0 + 1) * HID + col];
    }
    v8f c = {};
#pragma unroll
    for (int kb = 0; kb < 5; ++kb) {
      const int cb = kb & 1, nb = cb ^ 1;
      if (kb + 1 < 5) {
#pragma unroll
        for (int i = 0; i < 8; ++i) {
          const int k0 = (kb + 1) * 32 + i * 4 + lh2;
          bf[nb][i][0] = W1[(k0 + 0) * HID + col];
          bf[nb][i][1] = W1[(k0 + 1) * HID + col];
        }
      }
#pragma unroll
      for (int i = 0; i < 8; ++i) {
        const int k0 = kb * 32 + i * 4 + lh2;
        v2f a;
        const float* p0 = (k0     < HK) ? (X + (size_t)arow * HK + k0)
                                        : (state + (size_t)arow * Kd + (k0 - HK));
        const float* p1 = (k0 + 1 < HK) ? (X + (size_t)arow * HK + k0 + 1)
                                        : (state + (size_t)arow * Kd + (k0 + 1 - HK));
        a[0] = *p0;
        a[1] = *p1;
        c = WMMA_F32(a, bf[cb][i], c);
      }
    }
#pragma unroll
    for (int r = 0; r < 8; ++r) {
      const int lr = mt * 16 + r + (lhalf << 3);
      act1[lr][nt * 16 + l16] = fmaxf(c[r] + b1[nt * 16 + l16], 0.f);
    }
  }
  __syncthreads();

  // ---- layer 2: (64 x 256) @ W2 (256 x 256); K = 8 batches of 8 steps ----
  for (int nt = nt0; nt < nt0 + 8; ++nt) {
    const int col = nt * 16 + l16;
    v2f bf[2][8];
#pragma unroll
    for (int i = 0; i < 8; ++i) {
      const int k0 = i * 4 + lh2;
      bf[0][i][0] = W2[(k0 + 0) * HID + col];
      bf[0][i][1] = W2[(k0 + 1) * HID + col];
    }
    v8f c = {};
#pragma unroll
    for (int kb = 0; kb < 8; ++kb) {
      const int cb = kb & 1, nb = cb ^ 1;
      if (kb + 1 < 8) {
#pragma unroll
        for (int i = 0; i < 8; ++i) {
          const int k0 = (kb + 1) * 32 + i * 4 + lh2;
          bf[nb][i][0] = W2[(k0 + 0) * HID + col];
          bf[nb][i][1] = W2[(k0 + 1) * HID + col];
        }
      }
#pragma unroll
      for (int i = 0; i < 8; ++i) {
        const int k0 = kb * 32 + i * 4 + lh2;
        v2f a = *(const v2f*)&act1[mt * 16 + l16][k0];
        c = WMMA_F32(a, bf[cb][i], c);
      }
    }
#pragma unroll
    for (int r = 0; r < 8; ++r) {
      const int lr = mt * 16 + r + (lhalf << 3);
      act2[lr][nt * 16 + l16] = fmaxf(c[r] + b2[nt * 16 + l16], 0.f);
    }
  }
  __syncthreads();

  // ---- layer 3: 256 -> 1 dot per row ----
  if (tid < 64) {
    float s = 0.f;
    for (int cc = 0; cc < HID; ++cc) s += act2[tid][cc] * Wo[cc];
    out[row0 + tid] = s + bo[0];
  }
}

// ---------------------------------------------------------------------------
extern "C" void kernel_launch(void* const* d_in, const int* in_sizes, int n_in,
                              void* d_out, int out_size, void* d_ws, size_t ws_size,
                              hipStream_t stream) {
  const float* state = (const float*)d_in[0];
  const float* Wq    = (const float*)d_in[1];
  const float* Wk    = (const float*)d_in[2];
  const float* Wv    = (const float*)d_in[3];
  const float* W1    = (const float*)d_in[4];
  const float* b1    = (const float*)d_in[5];
  const float* W2    = (const float*)d_in[6];
  const float* b2    = (const float*)d_in[7];
  const float* Wo    = (const float*)d_in[8];
  const float* bo    = (const float*)d_in[9];
  float* out = (float*)d_out;

  float* ws = (float*)d_ws;
  const size_t n = (size_t)Bsz * Tn * HK;  // 524288 floats per buffer
  float* Q  = ws;
  float* Kf = ws + n;
  float* Vf = ws + 2 * n;
  float* X  = ws + 3 * n;

  proj_kernel<<<dim3(Bsz, 3), 256, 0, stream>>>(state, Wq, Wk, Wv, Q, Kf, Vf);
  attn_kernel<<<dim3(Bsz, Hn), 256, 0, stream>>>(Q, Kf, Vf, X);
  mlp_kernel<<<dim3((Bsz * Tn) / 64), 256, 0, stream>>>(X, state, W1, b1, W2, b2, Wo, bo, out);
}